// GCN_52304111730991
// MI455X (gfx1250) — compile-verified
//
#include <hip/hip_runtime.h>

typedef __attribute__((ext_vector_type(2))) float v2f;
typedef __attribute__((ext_vector_type(8))) float v8f;

#define N_NODES   100000
#define N_FEAT    128
#define HIDDEN    20
#define N_CLASSES 10
#define PAD1      32               // padded row stride for h0/h1 (128 B lines)
#define PAD2      16               // padded row stride for h2/h2a (64 B)
#define M_TILES   (N_NODES / 16)   // 6250, exact: no ragged M tiles

// ---------------- degree / normalization ----------------

__global__ void gcn_init_deg(float* deg) {
    int i = blockIdx.x * blockDim.x + threadIdx.x;
    if (i < N_NODES) deg[i] = 1.0f;            // self-loop contribution
}

__global__ void gcn_deg_accum(const int* __restrict__ dst, float* deg, int E) {
    int e = blockIdx.x * blockDim.x + threadIdx.x;
    if (e < E) atomicAdd(&deg[dst[e]], 1.0f);
}

__global__ void gcn_rsqrt(float* deg) {
    int i = blockIdx.x * blockDim.x + threadIdx.x;
    if (i < N_NODES) deg[i] = rsqrtf(deg[i]);  // deg >= 1 always (self loops)
}

// ---------------- GEMM1: h0 = x @ W1 (fp32 WMMA 16x16x4) + fused h1 init ----
// One wave per 16-row M tile. W1 staged in LDS pre-swizzled into fragment
// order Bf[k/2][col][2] (zero-padded cols 20..31), so each lane's B fragment
// is a single aligned ds_load_b64. Epilogue stores both h0 (row-padded to 32)
// and the self-loop accumulator init h1 = dis^2 * h0.

__global__ void gcn_gemm1(const float* __restrict__ x,
                          const float* __restrict__ W1,
                          const float* __restrict__ dis,
                          float* __restrict__ h0,
                          float* __restrict__ h1) {
    __shared__ float Bf[(N_FEAT / 2) * 32 * 2];    // 16 KB: [k/2][col][j]

    for (int idx = threadIdx.x; idx < (N_FEAT / 2) * 32 * 2; idx += blockDim.x) {
        const int p = idx >> 6;                    // k-pair
        const int c = (idx >> 1) & 31;             // column
        const int j = idx & 1;                     // element in pair
        Bf[idx] = (c < HIDDEN) ? W1[(2 * p + j) * HIDDEN + c] : 0.0f;
    }
    __syncthreads();

    const int lane = threadIdx.x & 31;
    const int wave = threadIdx.x >> 5;
    const int tile = blockIdx.x * (blockDim.x >> 5) + wave;
    if (tile >= M_TILES) return;                   // wave-uniform: EXEC all-1s
    const int row0 = tile << 4;
    const int half = lane >> 4;                    // 0: K={0,1}; 1: K={2,3}
    const int l16  = lane & 15;
    const float* arow = x + (size_t)(row0 + l16) * N_FEAT;

    v8f c0 = {};   // output cols 0..15
    v8f c1 = {};   // output cols 16..31 (cols 20..31 stay 0)

    #pragma unroll 4
    for (int k0 = 0; k0 < N_FEAT; k0 += 4) {
        const int kb = k0 + 2 * half;
        const int p = kb >> 1;
        v2f a  = *(const v2f*)(arow + kb);                      // global b64
        v2f b0 = *(const v2f*)&Bf[p * 64 + l16 * 2];            // ds b64
        v2f b1 = *(const v2f*)&Bf[p * 64 + (16 + l16) * 2];     // ds b64

        c0 = __builtin_amdgcn_wmma_f32_16x16x4_f32(false, a, false, b0, (short)0, c0, false, false);
        c1 = __builtin_amdgcn_wmma_f32_16x16x4_f32(false, a, false, b1, (short)0, c1, false, false);
    }

    // D layout: VGPR r -> row = row0 + r + 8*half, col = l16 (+16 for tile 1)
    #pragma unroll
    for (int r = 0; r < 8; ++r) {
        const int row = row0 + r + 8 * half;
        const float di = dis[row];                 // uniform per 16-lane half
        const float d2 = di * di;
        float* p0 = h0 + (size_t)row * PAD1;
        float* p1 = h1 + (size_t)row * PAD1;
        p0[l16]      = c0[r];
        p0[16 + l16] = c1[r];
        p1[l16]      = d2 * c0[r];                 // fused self-loop init
        p1[16 + l16] = d2 * c1[r];
    }
}

// ---------------- edge scatter: out[dst] += dis[src]*dis[dst]*h[src] --------
// Rows padded: PAD=32 -> gather is 5 x float4 from one 128B line;
// PAD=16 -> 2 x float4 + 1 x float2 from one line.

template <int F, int PAD>
__global__ void gcn_propagate(const int* __restrict__ src,
                              const int* __restrict__ dst,
                              const float* __restrict__ dis,
                              const float* __restrict__ h,
                              float* __restrict__ out, int E) {
    int e = blockIdx.x * blockDim.x + threadIdx.x;
    if (e >= E) return;
    const int s = src[e];
    const int d = dst[e];
    const float nrm = dis[s] * dis[d];
    const float* hrow = h + (size_t)s * PAD;
    float* orow = out + (size_t)d * PAD;

    float v[F];
    #pragma unroll
    for (int j = 0; j + 4 <= F; j += 4) {          // 16B-aligned chunks
        float4 t = *(const float4*)(hrow + j);
        v[j] = t.x; v[j + 1] = t.y; v[j + 2] = t.z; v[j + 3] = t.w;
    }
    if constexpr (F % 4 != 0) {                    // 8B tail (F=10)
        float2 t = *(const float2*)(hrow + (F & ~3));
        v[(F & ~3)] = t.x; v[(F & ~3) + 1] = t.y;
    }
    #pragma unroll
    for (int f = 0; f < F; ++f) atomicAdd(&orow[f], nrm * v[f]);
}

// ---------------- GEMM2: h2 = relu(h1+b1) @ W2, fused h2a init -------------
// W2 staged in LDS fragment order [k/2][col][2], zero-padded to 16 cols.

__global__ void gcn_gemm2(const float* __restrict__ h1,
                          const float* __restrict__ b1,
                          const float* __restrict__ W2,
                          const float* __restrict__ dis,
                          float* __restrict__ h2,
                          float* __restrict__ h2a) {
    __shared__ float Bf2[(HIDDEN / 2) * 16 * 2];   // 10 k-pairs x 16 cols x 2
    __shared__ float b1s[HIDDEN];

    for (int idx = threadIdx.x; idx < (HIDDEN / 2) * 16 * 2; idx += blockDim.x) {
        const int p = idx >> 5;
        const int c = (idx >> 1) & 15;
        const int j = idx & 1;
        Bf2[idx] = (c < N_CLASSES) ? W2[(2 * p + j) * N_CLASSES + c] : 0.0f;
    }
    if (threadIdx.x < HIDDEN) b1s[threadIdx.x] = b1[threadIdx.x];
    __syncthreads();

    const int lane = threadIdx.x & 31;
    const int wave = threadIdx.x >> 5;
    const int tile = blockIdx.x * (blockDim.x >> 5) + wave;
    if (tile >= M_TILES) return;                   // wave-uniform
    const int row0 = tile << 4;
    const int half = lane >> 4;
    const int l16  = lane & 15;
    const float* arow = h1 + (size_t)(row0 + l16) * PAD1;

    v8f c = {};
    #pragma unroll
    for (int k0 = 0; k0 < HIDDEN; k0 += 4) {       // 20 = 5 exact steps
        const int kb = k0 + 2 * half;
        v2f ar = *(const v2f*)(arow + kb);          // global b64, 8B aligned
        v2f a;                                      // fused bias + relu on A
        a.x = fmaxf(ar.x + b1s[kb],     0.0f);
        a.y = fmaxf(ar.y + b1s[kb + 1], 0.0f);
        v2f b = *(const v2f*)&Bf2[(kb >> 1) * 32 + l16 * 2];   // ds b64
        c = __builtin_amdgcn_wmma_f32_16x16x4_f32(false, a, false, b, (short)0, c, false, false);
    }

    #pragma unroll
    for (int r = 0; r < 8; ++r) {
        const int row = row0 + r + 8 * half;
        const float di = dis[row];
        const float d2 = di * di;
        h2 [(size_t)row * PAD2 + l16] = c[r];
        h2a[(size_t)row * PAD2 + l16] = d2 * c[r];  // fused self-loop init
    }
}

// ---------------- log_softmax(h2a + b2) ----------------

__global__ void gcn_logsoftmax(const float* __restrict__ h2a,
                               const float* __restrict__ b2,
                               float* __restrict__ out) {
    int i = blockIdx.x * blockDim.x + threadIdx.x;
    if (i >= N_NODES) return;
    const float* row = h2a + (size_t)i * PAD2;      // 64B-aligned row
    float z[N_CLASSES];
    {
        float4 t0 = *(const float4*)(row);
        float4 t1 = *(const float4*)(row + 4);
        float2 t2 = *(const float2*)(row + 8);
        z[0] = t0.x; z[1] = t0.y; z[2] = t0.z; z[3] = t0.w;
        z[4] = t1.x; z[5] = t1.y; z[6] = t1.z; z[7] = t1.w;
        z[8] = t2.x; z[9] = t2.y;
    }
    float m = -3.402823466e38f;
    #pragma unroll
    for (int c = 0; c < N_CLASSES; ++c) {
        z[c] += b2[c];
        m = fmaxf(m, z[c]);
    }
    float s = 0.0f;
    #pragma unroll
    for (int c = 0; c < N_CLASSES; ++c) s += __expf(z[c] - m);
    const float l = __logf(s);
    #pragma unroll
    for (int c = 0; c < N_CLASSES; ++c) out[(size_t)i * N_CLASSES + c] = z[c] - m - l;
}

// ---------------- launch ----------------

extern "C" void kernel_launch(void* const* d_in, const int* in_sizes, int n_in,
                              void* d_out, int out_size, void* d_ws, size_t ws_size,
                              hipStream_t stream) {
    const float* x  = (const float*)d_in[0];
    const int*   ei = (const int*)d_in[1];
    const float* W1 = (const float*)d_in[2];
    const float* b1 = (const float*)d_in[3];
    const float* W2 = (const float*)d_in[4];
    const float* b2 = (const float*)d_in[5];
    float* out = (float*)d_out;

    const int E = in_sizes[1] / 2;
    const int* src = ei;
    const int* dst = ei + E;

    // workspace layout (floats); all region bases 128B-aligned
    float* ws  = (float*)d_ws;
    float* dis = ws;                                    // 100000 (deg -> dis in place)
    float* h0  = ws + 102400;                           // N*32 padded
    float* h1  = h0 + (size_t)N_NODES * PAD1;           // N*32 padded
    float* h2  = h1 + (size_t)N_NODES * PAD1;           // N*16 padded
    float* h2a = h2 + (size_t)N_NODES * PAD2;           // N*16 padded

    const int TB = 256;
    const int nodeBlocks = (N_NODES + TB - 1) / TB;
    const int edgeBlocks = (E + TB - 1) / TB;
    const int gemmBlocks = (M_TILES + (TB / 32) - 1) / (TB / 32);

    // degree + normalization
    gcn_init_deg<<<nodeBlocks, TB, 0, stream>>>(dis);
    gcn_deg_accum<<<edgeBlocks, TB, 0, stream>>>(dst, dis, E);
    gcn_rsqrt<<<nodeBlocks, TB, 0, stream>>>(dis);

    // layer 1: GEMM (+fused self-loop init) then edge scatter
    gcn_gemm1<<<gemmBlocks, TB, 0, stream>>>(x, W1, dis, h0, h1);
    gcn_propagate<HIDDEN, PAD1><<<edgeBlocks, TB, 0, stream>>>(src, dst, dis, h0, h1, E);

    // layer 2: GEMM with fused bias+relu (+fused self-loop init), edge scatter
    gcn_gemm2<<<gemmBlocks, TB, 0, stream>>>(h1, b1, W2, dis, h2, h2a);
    gcn_propagate<N_CLASSES, PAD2><<<edgeBlocks, TB, 0, stream>>>(src, dst, dis, h2, h2a, E);

    // output
    gcn_logsoftmax<<<nodeBlocks, TB, 0, stream>>>(h2a, b2, out);
}